// Yolov2Loss_64201171141142
// MI455X (gfx1250) — compile-verified
//
#include <hip/hip_runtime.h>
#include <hip/hip_bf16.h>

// Problem constants (from reference setup_inputs)
#define B_N 32
#define S_N 52
#define A_N 5
#define C_N 80
#define T_N 50
#define PD_N (A_N * (5 + C_N))          // 425 channels per pixel
#define NTGT (B_N * T_N)                // 1600 targets
#define NCELL (B_N * A_N * S_N * S_N)   // 432640 cells
#define TPB 256
#define NBLK 640

typedef __attribute__((ext_vector_type(2))) float v2f_t;
typedef __attribute__((ext_vector_type(8))) float v8f_t;

__constant__ float c_aw[A_N] = {1.3221f, 3.19275f, 5.05587f, 9.47112f, 11.2364f};
__constant__ float c_ah[A_N] = {1.73145f, 4.00944f, 8.09892f, 4.84053f, 10.0071f};

// Full-wave (32-lane) sum using V_WMMA_F32_16X16X4_F32.
// A: lane m holds A[m][0]=x (VGPR0) and A[m][1]=0 (VGPR1); lanes 16..31 map to K=2,3.
// B = all ones  =>  D[m][n] = x(m) + x(m+16) for every n.
// C/D layout: lane n (n<16) VGPR r = D[r][n]; lane n+16 VGPR r = D[r+8][n].
// Sum of the 8 D VGPRs per lane + shfl_xor(16) => total in every lane.
// Requires EXEC all-ones: callers invoke this only at fully-converged points.
__device__ __forceinline__ float wave_sum_wmma(float x) {
  v2f_t a = {x, 0.0f};
  v2f_t b = {1.0f, 1.0f};
  v8f_t c = {0.f, 0.f, 0.f, 0.f, 0.f, 0.f, 0.f, 0.f};
  c = __builtin_amdgcn_wmma_f32_16x16x4_f32(false, a, false, b, (short)0, c,
                                            false, false);
  float s = c[0] + c[1] + c[2] + c[3] + c[4] + c[5] + c[6] + c[7];
  s += __shfl_xor(s, 16, 32);
  return s;
}

__device__ __forceinline__ float sigmoidf(float x) {
  return 1.0f / (1.0f + expf(-x));
}

// ---------------- Kernel 1: init owner map ----------------
__global__ void yolo_init_owner(int* __restrict__ owner) {
  int i = blockIdx.x * blockDim.x + threadIdx.x;
  if (i < NCELL) owner[i] = -1;
}

// ---------------- Kernel 2: scatter targets -> owner cells ----------------
// Deterministic duplicate resolution: highest target index wins (atomicMax).
__global__ void yolo_scatter(const float* __restrict__ tgt,
                             int* __restrict__ owner) {
  int t = blockIdx.x * blockDim.x + threadIdx.x;
  if (t >= NTGT) return;
  int b = t / T_N;
  const float* tp = tgt + (size_t)t * 5;
  float gx = tp[0], gy = tp[1], gw = tp[2], gh = tp[3];
  float gtw = gw * (float)S_N;
  float gth = gh * (float)S_N;

  // anchor wh-IoU argmax (first-max wins, matching jnp.argmax)
  int best = 0;
  float bestv = -1.0f;
#pragma unroll
  for (int a = 0; a < A_N; ++a) {
    float aw = c_aw[a], ah = c_ah[a];
    float inter = fminf(gtw, aw) * fminf(gth, ah);
    float uni = gtw * gth + aw * ah - inter;
    float iou = uni > 0.0f ? inter / uni : 0.0f;
    if (iou > bestv) { bestv = iou; best = a; }
  }
  int gi = (int)(gx * (float)S_N);   // gx in [0,0.999) -> gi in [0,51]
  int gj = (int)(gy * (float)S_N);
  int cell = ((b * A_N + best) * S_N + gj) * S_N + gi;
  atomicMax(&owner[cell], t);
}

// ---------------- Kernel 3: streaming reduction over all cells ----------------
// Only touches the objectness channel per cell (plus tx..th at owned cells).
__global__ void yolo_reduce(const float* __restrict__ pred,
                            const float* __restrict__ tgt,
                            const int* __restrict__ owner,
                            float* __restrict__ partials) {
  float Sc = 0.0f;  // raw coord offset+scale sum (no lambdas)
  float So = 0.0f;  // sum (iou - conf)^2 at object cells
  float Sn = 0.0f;  // sum conf^2 at no-object cells
  const int stride = gridDim.x * blockDim.x;

  for (int i = blockIdx.x * blockDim.x + threadIdx.x; i < NCELL; i += stride) {
    int x = i % S_N;
    int y = (i / S_N) % S_N;
    int a = (i / (S_N * S_N)) % A_N;
    int b = i / (S_N * S_N * A_N);
    int base = ((b * S_N + y) * S_N + x) * PD_N + a * (5 + C_N);

    float score = pred[base + 4];
    // Pull the next anchor's objectness cacheline while sigmoid executes
    // (speculative prefetch; OOB at array tail is silently dropped).
    __builtin_prefetch(&pred[base + 4 + (5 + C_N)], 0, 0);

    float conf = sigmoidf(score);
    int t = owner[i];
    if (t >= 0) {
      const float* tp = tgt + (size_t)t * 5;
      float gx = tp[0], gy = tp[1], gw = tp[2], gh = tp[3];
      float gtw = gw * (float)S_N;
      float gth = gh * (float)S_N;
      float ggx = gx * (float)S_N - (float)x;  // gi == x, gj == y by scatter
      float ggy = gy * (float)S_N - (float)y;

      float tx = pred[base + 0];
      float ty = pred[base + 1];
      float tw = pred[base + 2];
      float th = pred[base + 3];
      float pbx = sigmoidf(tx);
      float pby = sigmoidf(ty);
      float pbw = expf(tw) * c_aw[a];
      float pbh = expf(th) * c_ah[a];

      // IoU in cxcywh form (offsets +gi/+gj kept to stay faithful)
      float cx1 = ggx + (float)x, cy1 = ggy + (float)y;
      float cx2 = pbx + (float)x, cy2 = pby + (float)y;
      float ix = fmaxf(0.0f, fminf(cx1 + 0.5f * gtw, cx2 + 0.5f * pbw) -
                            fmaxf(cx1 - 0.5f * gtw, cx2 - 0.5f * pbw));
      float iy = fmaxf(0.0f, fminf(cy1 + 0.5f * gth, cy2 + 0.5f * pbh) -
                            fmaxf(cy1 - 0.5f * gth, cy2 - 0.5f * pbh));
      float inter = ix * iy;
      float uni = gtw * gth + pbw * pbh - inter;
      float iou = uni > 0.0f ? inter / uni : 0.0f;

      const float eps = 1e-6f;
      float dox = pbx - ggx, doy = pby - ggy;
      float dsw = sqrtf(pbw + eps) - sqrtf(gtw + eps);
      float dsh = sqrtf(pbh + eps) - sqrtf(gth + eps);
      Sc += dox * dox + doy * doy + dsw * dsw + dsh * dsh;
      float de = iou - conf;
      So += de * de;
    } else {
      Sn += conf * conf;
    }
  }

  // Fully converged here: WMMA wave reductions (EXEC all-ones).
  Sc = wave_sum_wmma(Sc);
  So = wave_sum_wmma(So);
  Sn = wave_sum_wmma(Sn);

  __shared__ float sm[3][TPB / 32];
  int wave = threadIdx.x >> 5;
  int lane = threadIdx.x & 31;
  if (lane == 0) {
    sm[0][wave] = Sc;
    sm[1][wave] = So;
    sm[2][wave] = Sn;
  }
  __syncthreads();
  if (threadIdx.x == 0) {
    float a0 = 0.f, a1 = 0.f, a2 = 0.f;
#pragma unroll
    for (int w = 0; w < TPB / 32; ++w) {
      a0 += sm[0][w];
      a1 += sm[1][w];
      a2 += sm[2][w];
    }
    partials[(size_t)blockIdx.x * 3 + 0] = a0;
    partials[(size_t)blockIdx.x * 3 + 1] = a1;
    partials[(size_t)blockIdx.x * 3 + 2] = a2;
  }
}

// ---------------- Kernel 4: finalize (single wave32) ----------------
__global__ void yolo_finalize(const float* __restrict__ partials,
                              float* __restrict__ out) {
  float a0 = 0.f, a1 = 0.f, a2 = 0.f;
  for (int i = threadIdx.x; i < NBLK; i += 32) {
    a0 += partials[(size_t)i * 3 + 0];
    a1 += partials[(size_t)i * 3 + 1];
    a2 += partials[(size_t)i * 3 + 2];
  }
  a0 = wave_sum_wmma(a0);
  a1 = wave_sum_wmma(a1);
  a2 = wave_sum_wmma(a2);
  if (threadIdx.x == 0) {
    // Reference double-applies lambdas in `total`:
    // total = 5*(5*Sc) + So + 0.5*(0.5*Sn); out1 = 5*Sc; out2 = So + 0.5*Sn
    out[0] = 25.0f * a0 + a1 + 0.25f * a2;
    out[1] = 5.0f * a0;
    out[2] = a1 + 0.5f * a2;
  }
}

extern "C" void kernel_launch(void* const* d_in, const int* in_sizes, int n_in,
                              void* d_out, int out_size, void* d_ws,
                              size_t ws_size, hipStream_t stream) {
  const float* pred = (const float*)d_in[0];  // [32,52,52,425] fp32
  const float* tgt  = (const float*)d_in[1];  // [32,50,5] fp32
  float* out = (float*)d_out;                 // 3 fp32 scalars

  // Workspace layout: owner map then block partials (offset is 256B aligned).
  int* owner = (int*)d_ws;                                     // NCELL ints
  float* partials = (float*)((char*)d_ws + (size_t)NCELL * 4); // 3*NBLK floats

  yolo_init_owner<<<(NCELL + TPB - 1) / TPB, TPB, 0, stream>>>(owner);
  yolo_scatter<<<(NTGT + TPB - 1) / TPB, TPB, 0, stream>>>(tgt, owner);
  yolo_reduce<<<NBLK, TPB, 0, stream>>>(pred, tgt, owner, partials);
  yolo_finalize<<<1, 32, 0, stream>>>(partials, out);
  (void)in_sizes; (void)n_in; (void)out_size; (void)ws_size;
}